// ProtoNetLayer_24017457119656
// MI455X (gfx1250) — compile-verified
//
#include <hip/hip_runtime.h>
#include <hip/hip_bf16.h>
#include <math.h>

// ProtoNet layer for MI455X (gfx1250, wave32).
// out[b,n,c] = -|| (X[b,n]-mean)/max(||X[b,n]-mean||,eps) - P[c] ||
// Expanded: q = s*r^2 + ||P_c||^2 - 2*r*dot_c ; out = -sqrt(max(q,0))
//   s = ||x-m||^2, r = 1/max(sqrt(s), eps), dot_c = (x-m).P_c
// dot_c computed with V_WMMA_F32_16X16X4_F32 (f32 tensor op), s/pp with VALU.

typedef float v2f __attribute__((ext_vector_type(2)));
typedef float v4f __attribute__((ext_vector_type(4)));
typedef float v8f __attribute__((ext_vector_type(8)));

#define PN_D        1024
#define PN_C        7
#define PN_TILE_M   16
#define PN_WAVES    4            // waves per block (wave32 -> 128 threads)
#define PN_LDSROWS  8            // rows 0-6: prototypes, row 7: mean
#define PN_LDSSTR   1036         // floats; 1036 % 64 == 12 -> conflict-free b128 reads
#define PN_EPS      1e-12f

__global__ __launch_bounds__(PN_WAVES * 32)
void protonet_l2_kernel(const float* __restrict__ X,
                        const float* __restrict__ P,
                        const float* __restrict__ mean,
                        float* __restrict__ out)
{
    __shared__ float smem[PN_LDSROWS * PN_LDSSTR];   // 33,152 bytes

    const int t = threadIdx.x;

    // ---- Stage prototypes (rows 0-6) and mean (row 7) into LDS ----
    for (int idx = t; idx < PN_LDSROWS * PN_D; idx += blockDim.x) {
        const int rr = idx >> 10;          // /1024
        const int cc = idx & (PN_D - 1);
        const float v = (rr < PN_C) ? P[rr * PN_D + cc] : mean[cc];
        smem[rr * PN_LDSSTR + cc] = v;
    }
    __syncthreads();

    const int wave = t >> 5;               // wave32
    const int lane = t & 31;
    const int h    = lane >> 4;            // k-half: 0 or 1
    const int mrow = lane & 15;            // A row within tile; also output column c
    const int tile = blockIdx.x * PN_WAVES + wave;
    const int row0 = tile * PN_TILE_M;

    const float* xrow = X + (size_t)(row0 + mrow) * PN_D;        // this lane's A row
    // B row: prototype (or mean) for column n = mrow; rows >= 8 alias 0-7 (outputs ignored)
    const float* brow = &smem[(mrow & 7) * PN_LDSSTR];
    const float* mrw  = &smem[7 * PN_LDSSTR];

    v8f   acc     = {};     // 16x16 f32 accumulator: D[g][lane] = dot(row g+8h, proto mrow)
    float s_part  = 0.0f;   // partial ||x-m||^2 for row mrow (this lane's k-half)
    float pp_part = 0.0f;   // partial ||P_n||^2 (this lane's k-half)

    // ---- Stream D in chunks of 8 k-values: 1 b128 global + 2 b128 LDS + 2 WMMAs ----
    for (int base = 0; base < PN_D; base += 8) {
        const int off = base + 4 * h;      // lane-half physical-k mapping (A and B agree)
        v4f xa = *(const v4f*)(xrow + off);
        v4f bb = *(const v4f*)(brow + off);
        v4f mm = *(const v4f*)(mrw  + off);

        __builtin_prefetch(xrow + off + 256, 0, 3);   // ~1KB ahead in this row's stream

        v4f a;
        a.x = xa.x - mm.x; a.y = xa.y - mm.y; a.z = xa.z - mm.z; a.w = xa.w - mm.w;

        s_part  = fmaf(a.x, a.x, fmaf(a.y, a.y, fmaf(a.z, a.z, fmaf(a.w, a.w, s_part))));
        pp_part = fmaf(bb.x, bb.x, fmaf(bb.y, bb.y, fmaf(bb.z, bb.z, fmaf(bb.w, bb.w, pp_part))));

        // Physical k slots per lane: (V0,V1) = (off, off+1) then (off+2, off+3).
        // B packed identically, so the contraction sums the same physical k set.
        v2f a0 = { a.x,  a.y  }, a1 = { a.z,  a.w  };
        v2f b0 = { bb.x, bb.y }, b1 = { bb.z, bb.w };

        acc = __builtin_amdgcn_wmma_f32_16x16x4_f32(false, a0, false, b0,
                                                    (short)0, acc, false, false);
        acc = __builtin_amdgcn_wmma_f32_16x16x4_f32(false, a1, false, b1,
                                                    (short)0, acc, false, false);
    }

    // ---- Complete cross-half reductions (lanes L and L+16 hold complementary k) ----
    float s_full = s_part  + __shfl_xor(s_part, 16, 32);   // valid on lanes mrow & mrow+16
    float pp     = pp_part + __shfl_xor(pp_part, 16, 32);  // ||P_c||^2 for c = mrow

    // ---- Final distance math; lane holds column c=mrow for rows g+8h ----
    float res[8];
    #pragma unroll
    for (int g = 0; g < 8; ++g) {
        const float s    = __shfl(s_full, g + 8 * h, 32);  // row (g+8h)'s norm^2
        const float nrm  = sqrtf(s);
        const float r    = 1.0f / fmaxf(nrm, PN_EPS);
        const float q    = fmaf(s, r * r, pp) - 2.0f * r * acc[g];
        res[g] = -sqrtf(fmaxf(q, 0.0f));
    }

    if (mrow < PN_C) {
        #pragma unroll
        for (int g = 0; g < 8; ++g) {
            const int row = row0 + g + 8 * h;
            out[(size_t)row * PN_C + mrow] = res[g];
        }
    }
}

extern "C" void kernel_launch(void* const* d_in, const int* in_sizes, int n_in,
                              void* d_out, int out_size, void* d_ws, size_t ws_size,
                              hipStream_t stream) {
    const float* X    = (const float*)d_in[0];   // [8,4096,1024] f32
    const float* P    = (const float*)d_in[1];   // [7,1024] f32
    const float* mean = (const float*)d_in[2];   // [1024] f32
    float* out        = (float*)d_out;           // [8,4096,7] f32

    const int rows   = in_sizes[0] / PN_D;            // 32768
    const int tiles  = rows / PN_TILE_M;              // 2048
    const int blocks = tiles / PN_WAVES;              // 512

    protonet_l2_kernel<<<blocks, PN_WAVES * 32, 0, stream>>>(X, P, mean, out);
}